// Linear_act_sp_55886114456225
// MI455X (gfx1250) — compile-verified
//
#include <hip/hip_runtime.h>
#include <stdint.h>

typedef __attribute__((ext_vector_type(16))) __bf16 v16bf;
typedef __attribute__((ext_vector_type(32))) __bf16 v32bf;
typedef __attribute__((ext_vector_type(8)))  float  v8f;

#define K_DIM 4096
#define N_DIM 4096
#define BM 128
#define BN 256
#define BKS 64            // original-K per SWMMAC step
#define LDAS 40           // stored-A LDS row stride (32 + 8 pad), ushorts
#define LDBS 72           // B LDS row stride (64 + 8 pad), ushorts

__device__ __forceinline__ unsigned short f2bf(float f) {
    // round-to-nearest-even fp32 -> bf16
    uint32_t u = __float_as_uint(f);
    uint32_t r = u + 0x7FFFu + ((u >> 16) & 1u);
    return (unsigned short)(r >> 16);
}

// Low 32 bits of a generic pointer to __shared__ == AS(3) byte offset.
__device__ __forceinline__ unsigned lds_off(const void* p) {
    return (unsigned)(unsigned long long)p;
}

// gfx1250 async global -> LDS copies (ASYNCcnt-tracked, no VGPR staging)
__device__ __forceinline__ void async_ld_b128(unsigned ldsOff, const void* g) {
    asm volatile("global_load_async_to_lds_b128 %0, %1, off"
                 :: "v"(ldsOff), "v"(g) : "memory");
}
__device__ __forceinline__ void async_ld_b32(unsigned ldsOff, const void* g) {
    asm volatile("global_load_async_to_lds_b32 %0, %1, off"
                 :: "v"(ldsOff), "v"(g) : "memory");
}
__device__ __forceinline__ void wait_async0() {
    asm volatile("s_wait_asynccnt 0x0" ::: "memory");
}

// ---------------------------------------------------------------------------
// Kernel 1: 2:4 magnitude prune -> SWMMAC operands.
// Each thread handles 2 contiguous groups of 4 (8 floats = 32B):
//   Ap   : the 2 kept values per group, ascending index order (4 bf16 = 8B)
//   AIdx : one byte = two nibbles, nibble = idx0 | (idx1<<2), idx0 < idx1
// Tie-break: lower index wins (matches jax top_k).
// ---------------------------------------------------------------------------
__global__ void __launch_bounds__(256) prune_pack_x(
    const float* __restrict__ x, unsigned short* __restrict__ Ap,
    unsigned char* __restrict__ AIdx, int npairs) {
    int t = blockIdx.x * blockDim.x + threadIdx.x;
    if (t >= npairs) return;
    const float4* xg = (const float4*)x;
    unsigned short vals[4];
    unsigned char nib[2];
#pragma unroll
    for (int h = 0; h < 2; ++h) {
        float4 v = xg[2 * t + h];
        float a[4]  = {v.x, v.y, v.z, v.w};
        float ax[4] = {fabsf(v.x), fabsf(v.y), fabsf(v.z), fabsf(v.w)};
        int i0 = -1, i1 = -1;
#pragma unroll
        for (int i = 0; i < 4; ++i) {
            int rank = 0;
#pragma unroll
            for (int j = 0; j < 4; ++j) {
                if (j == i) continue;
                rank += (ax[j] > ax[i]) || ((ax[j] == ax[i]) && (j < i));
            }
            if (rank < 2) { if (i0 < 0) i0 = i; else i1 = i; }
        }
        vals[2 * h]     = f2bf(a[i0]);
        vals[2 * h + 1] = f2bf(a[i1]);
        nib[h] = (unsigned char)(i0 | (i1 << 2));
    }
    uint2 p;
    p.x = (uint32_t)vals[0] | ((uint32_t)vals[1] << 16);
    p.y = (uint32_t)vals[2] | ((uint32_t)vals[3] << 16);
    ((uint2*)Ap)[t] = p;
    AIdx[t] = (unsigned char)(nib[0] | (nib[1] << 4));
}

// ---------------------------------------------------------------------------
// Kernel 2: weight fp32 -> bf16 ([N,K], K-major; 32 MB -> L2-resident)
// ---------------------------------------------------------------------------
__global__ void __launch_bounds__(256) convert_w_to_bf16(
    const float* __restrict__ w, unsigned short* __restrict__ wb, int ngroups) {
    int g = blockIdx.x * blockDim.x + threadIdx.x;
    if (g >= ngroups) return;
    const float4 v = ((const float4*)w)[g];
    uint2 p;
    p.x = (uint32_t)f2bf(v.x) | ((uint32_t)f2bf(v.y) << 16);
    p.y = (uint32_t)f2bf(v.z) | ((uint32_t)f2bf(v.w) << 16);
    ((uint2*)wb)[g] = p;
}

// ---------------------------------------------------------------------------
// Kernel 3: 2:4 sparse bf16 GEMM via v_swmmac_f32_16x16x64_bf16 with
// double-buffered async global->LDS staging.
// C[M,N] = expand(Ap,AIdx)[M,K] * W[N,K]^T, fp32 accumulate.
// Block: 256 threads = 8 waves (2x4); wave tile 64x64 = 4x4 subtiles.
// Pipeline: wait_asynccnt(tile t) -> barrier -> issue tile t+1 (other
// buffer, overlaps compute) -> ds_load frags + 16 swmmac on tile t.
// ---------------------------------------------------------------------------
__global__ void __launch_bounds__(256) gemm_bf16_swmmac(
    const unsigned short* __restrict__ Ap,   // [M, K/2] packed bf16
    const unsigned int*  __restrict__ AIdx,  // [M, K/32] u32 of nibbles
    const unsigned short* __restrict__ W,    // [N, K] bf16
    float* __restrict__ C, int M) {
    __shared__ unsigned short As[2][BM * LDAS];   // 2 x 10 KB
    __shared__ unsigned int   Is[2][BM * 2];      // 2 x  1 KB
    __shared__ unsigned short Bs[2][BN * LDBS];   // 2 x 36 KB

    const int tid   = threadIdx.x;
    const int lane  = tid & 31;
    const int wave  = tid >> 5;
    const int waveM = wave >> 2;        // 0..1
    const int waveN = wave & 3;         // 0..3
    const int bm    = blockIdx.y * BM;
    const int bn    = blockIdx.x * BN;
    const int half  = lane >> 4;
    const int l15   = lane & 15;

    const unsigned asOff[2] = {lds_off(&As[0][0]), lds_off(&As[1][0])};
    const unsigned isOff[2] = {lds_off(&Is[0][0]), lds_off(&Is[1][0])};
    const unsigned bsOff[2] = {lds_off(&Bs[0][0]), lds_off(&Bs[1][0])};

    auto issue_tile = [&](int buf, int k0) {
        // stored-A tile: 128 rows x 32 stored ushorts = 512 x 16B chunks
#pragma unroll
        for (int u = 0; u < 2; ++u) {
            int c = tid + u * 256;
            int row = c >> 2, seg = (c & 3) * 8;
            async_ld_b128(asOff[buf] + (unsigned)(row * LDAS + seg) * 2,
                          Ap + (size_t)(bm + row) * (K_DIM / 2) + (k0 >> 1) + seg);
        }
        // index tile: 128 rows x 2 u32 (1 per thread)
        {
            int row = tid >> 1, w = tid & 1;
            async_ld_b32(isOff[buf] + (unsigned)(row * 2 + w) * 4,
                         AIdx + (size_t)(bm + row) * (K_DIM / 32) + (k0 >> 5) + w);
        }
        // B tile: 256 rows x 64 ushorts = 2048 x 16B chunks
#pragma unroll
        for (int u = 0; u < 8; ++u) {
            int c = tid + u * 256;
            int row = c >> 3, seg = (c & 7) * 8;
            async_ld_b128(bsOff[buf] + (unsigned)(row * LDBS + seg) * 2,
                          W + (size_t)(bn + row) * K_DIM + k0 + seg);
        }
    };

    v8f acc[4][4];
#pragma unroll
    for (int i = 0; i < 4; ++i)
#pragma unroll
        for (int j = 0; j < 4; ++j) acc[i][j] = (v8f)(0.0f);

    union FragA { v16bf v; uint4 q[2]; };
    union FragB { v32bf v; uint4 q[4]; };

    issue_tile(0, 0);
    int buf = 0;
    for (int k0 = 0; k0 < K_DIM; k0 += BKS) {
        wait_async0();        // this wave's share of tile t has landed in LDS
        __syncthreads();      // all waves' shares landed; prev buffer free
        if (k0 + BKS < K_DIM) issue_tile(buf ^ 1, k0 + BKS);  // overlap

        FragA aF[4];
        int   idxv[4];
        FragB bF[4];
#pragma unroll
        for (int i = 0; i < 4; ++i) {
            int r = waveM * 64 + i * 16 + l15;
            int base = r * LDAS + half * 16;
            aF[i].q[0] = *(const uint4*)(&As[buf][base]);
            aF[i].q[1] = *(const uint4*)(&As[buf][base + 8]);
            idxv[i]    = (int)Is[buf][r * 2 + half];
        }
#pragma unroll
        for (int j = 0; j < 4; ++j) {
            int rb = (waveN * 64 + j * 16 + l15) * LDBS;
            bF[j].q[0] = *(const uint4*)(&Bs[buf][rb + half * 16]);
            bF[j].q[1] = *(const uint4*)(&Bs[buf][rb + half * 16 + 8]);
            bF[j].q[2] = *(const uint4*)(&Bs[buf][rb + 32 + half * 16]);
            bF[j].q[3] = *(const uint4*)(&Bs[buf][rb + 32 + half * 16 + 8]);
        }
#pragma unroll
        for (int i = 0; i < 4; ++i)
#pragma unroll
            for (int j = 0; j < 4; ++j)
                acc[i][j] = __builtin_amdgcn_swmmac_f32_16x16x64_bf16(
                    /*neg_a=*/false, aF[i].v, /*neg_b=*/false, bF[j].v,
                    acc[i][j], idxv[i],
                    /*reuse_a=*/false, /*reuse_b=*/false);
        buf ^= 1;
    }

    // Epilogue: C/D layout -> VGPR r holds M = r + 8*half, N = lane&15
#pragma unroll
    for (int i = 0; i < 4; ++i)
#pragma unroll
        for (int j = 0; j < 4; ++j)
#pragma unroll
            for (int r = 0; r < 8; ++r) {
                int m = bm + waveM * 64 + i * 16 + r + half * 8;
                int n = bn + waveN * 64 + j * 16 + l15;
                C[(size_t)m * N_DIM + n] = acc[i][j][r];
            }
}

// ---------------------------------------------------------------------------
extern "C" void kernel_launch(void* const* d_in, const int* in_sizes, int n_in,
                              void* d_out, int out_size, void* d_ws, size_t ws_size,
                              hipStream_t stream) {
    const float* x = (const float*)d_in[0];     // [bs, seq, 4096] fp32
    const float* w = (const float*)d_in[1];     // [4096, 4096] fp32
    float* out = (float*)d_out;                 // [bs, seq, 4096] fp32

    const int K = K_DIM, N = N_DIM;
    const int M = in_sizes[0] / K;              // 8192

    unsigned short* Ap  = (unsigned short*)d_ws;            // 32 MB packed A
    unsigned int*  AIdx = (unsigned int*)(Ap + (size_t)M * K / 2);  // 4 MB
    unsigned short* Wb  = (unsigned short*)(AIdx + (size_t)M * K / 32); // 32 MB

    int npairs = (M * K) / 8;                   // 2 groups per thread
    prune_pack_x<<<(npairs + 255) / 256, 256, 0, stream>>>(
        x, Ap, (unsigned char*)AIdx, npairs);
    int wg = (N * K) / 4;
    convert_w_to_bf16<<<(wg + 255) / 256, 256, 0, stream>>>(w, Wb, wg);

    dim3 grid(N / BN, M / BM);   // 16 x 64
    gemm_bf16_swmmac<<<grid, 256, 0, stream>>>(Ap, AIdx, Wb, out, M);
}